// SpatialProductLayer_75737453298220
// MI455X (gfx1250) — compile-verified
//
#include <hip/hip_runtime.h>

// Problem constants (from the reference)
#define IC    64      // input channels
#define LIN   8192    // input length
#define OC    256     // output channels
#define KS    4       // kernel taps
#define TOUT  4099    // output length: (8192 + 12 - 7)/2 + 1
#define NB    32      // batch
#define TT    128     // output t-values per block
#define NJ    (TT + KS - 1)   // 131 staged even positions per channel row
#define PITCH 132             // LDS row pitch (floats)

// ---------------------------------------------------------------------------
// Kernel 1: decode the one-hot weights into idx[oc][k] = argmax_c w[oc][c][k]
// weight layout: [OC][IC][KS] float32
// ---------------------------------------------------------------------------
__global__ __launch_bounds__(256) void decode_idx_kernel(const float* __restrict__ w,
                                                         int* __restrict__ idx) {
    int g = blockIdx.x * 256 + threadIdx.x;        // 0..1023 == oc*4 + k
    if (g >= OC * KS) return;
    int oc = g >> 2;
    int k  = g & 3;
    int c_found = 0;
    for (int c = 0; c < IC; ++c) {
        if (w[(oc * IC + c) * KS + k] > 0.5f) c_found = c;
    }
    idx[g] = c_found;
}

// ---------------------------------------------------------------------------
// Kernel 2: sparse gather "conv".
//   out[b,oc,t0+i] = sum_k xe[idx[oc][k]][ i + k ]
// where xe[c][j] = x[b, c, 2*t0 - 6 + 2*j]  (zero outside [0, LIN)).
// Staging uses CDNA5 async global->LDS loads (ASYNCcnt) with NT temporal
// hint to compact the stride-2 (even-position) stream into a dense LDS slab
// without a VGPR round-trip and without polluting L2 (zero reuse stream).
// ---------------------------------------------------------------------------
__global__ __launch_bounds__(256) void conv_gather_kernel(const float* __restrict__ x,
                                                          const int* __restrict__ idx,
                                                          float* __restrict__ out) {
    __shared__ float xe[IC * PITCH];     // ~33.8 KB
    __shared__ int   idxs[OC * KS];      // 4 KB

    const int tid = threadIdx.x;
    const int t0  = blockIdx.x * TT;
    const int b   = blockIdx.y;

    // Stage the decoded indices (256 threads x int4 = 1024 ints).
    ((int4*)idxs)[tid] = ((const int4*)idx)[tid];

    // Stage even-position input slab: xe[c][j] = x[b][c][2*t0 - 6 + 2*j]
    const unsigned long long xbase =
        (unsigned long long)(const void*)(x + (size_t)b * IC * LIN);
    const int p0 = 2 * t0 - 6;
    for (int e = tid; e < IC * NJ; e += 256) {
        int c   = e / NJ;
        int j   = e - c * NJ;
        int pos = p0 + 2 * j;
        unsigned lds_off = (unsigned)(size_t)(&xe[c * PITCH + j]);
        if (pos >= 0 && pos < LIN) {
            unsigned voff = (unsigned)((c * LIN + pos) * 4);
            // CDNA5 async copy: copies 4 bytes from the global address xbase plus voff to the LDS address in lds_off (per lane,
            // EXEC-predicated), tracked by ASYNCcnt. NT: single-use stream.
            asm volatile("global_load_async_to_lds_b32 %0, %1, %2 th:TH_LOAD_NT"
                         :
                         : "v"(lds_off), "v"(voff), "s"(xbase)
                         : "memory");
        } else {
            xe[c * PITCH + j] = 0.0f;    // zero padding outside the tensor
        }
    }
    asm volatile("s_wait_asynccnt 0x0" ::: "memory");
    __syncthreads();

    // Compute: wave wv handles oc = wv + 8*m; lane L handles t = t0 + L + 32*j.
    // LDS reads: consecutive lanes -> consecutive banks (conflict-free).
    // Global stores: consecutive lanes -> consecutive floats (coalesced), NT.
    const int wv = tid >> 5;
    const int L  = tid & 31;
    float* __restrict__ outb = out + (size_t)b * OC * TOUT;
    const bool full_tile = (t0 + TT) <= TOUT;

    for (int m = 0; m < 32; ++m) {
        const int oc = (m << 3) | wv;
        const int4 c4 = ((const int4*)idxs)[oc];       // one ds_load_b128
        const float* __restrict__ r0 = xe + c4.x * PITCH;
        const float* __restrict__ r1 = xe + c4.y * PITCH + 1;
        const float* __restrict__ r2 = xe + c4.z * PITCH + 2;
        const float* __restrict__ r3 = xe + c4.w * PITCH + 3;
        float* __restrict__ orow = outb + (size_t)oc * TOUT + t0;
        if (full_tile) {
#pragma unroll
            for (int j = 0; j < 4; ++j) {
                const int i = L + (j << 5);            // 0..127
                const float v = r0[i] + r1[i] + r2[i] + r3[i];
                __builtin_nontemporal_store(v, &orow[i]);
            }
        } else {
#pragma unroll
            for (int j = 0; j < 4; ++j) {
                const int i = L + (j << 5);
                if (t0 + i < TOUT) {
                    const float v = r0[i] + r1[i] + r2[i] + r3[i];
                    __builtin_nontemporal_store(v, &orow[i]);
                }
            }
        }
    }
}

// ---------------------------------------------------------------------------
// Launch: decode indices into d_ws, then the gather kernel.
// d_in[0] = x [32*64*8192] f32, d_in[1] = weight [256*64*4] f32
// d_out   = [32*256*4099] f32
// ---------------------------------------------------------------------------
extern "C" void kernel_launch(void* const* d_in, const int* in_sizes, int n_in,
                              void* d_out, int out_size, void* d_ws, size_t ws_size,
                              hipStream_t stream) {
    (void)in_sizes; (void)n_in; (void)out_size; (void)ws_size;
    const float* x   = (const float*)d_in[0];
    const float* wgt = (const float*)d_in[1];
    int*   idxp = (int*)d_ws;           // 1024 ints = 4 KB scratch
    float* out  = (float*)d_out;

    decode_idx_kernel<<<dim3((OC * KS + 255) / 256), dim3(256), 0, stream>>>(wgt, idxp);

    dim3 grid((TOUT + TT - 1) / TT, NB);   // (33, 32)
    conv_gather_kernel<<<grid, dim3(256), 0, stream>>>(x, idxp, out);
}